// Pointnet2Backbone_adaptive_33672543600992
// MI455X (gfx1250) — compile-verified
//
#include <hip/hip_runtime.h>

typedef float v2f __attribute__((ext_vector_type(2)));
typedef float v8f __attribute__((ext_vector_type(8)));

#define TPB 256

// ---------------------------------------------------------------- prep
__global__ void prep_kernel(const float* __restrict__ pc, const float* __restrict__ hmi,
                            float* __restrict__ xyz0, float* __restrict__ feat0,
                            float* __restrict__ hm0, int N) {
    int i = blockIdx.x * blockDim.x + threadIdx.x;
    if (i >= N) return;
    xyz0[i * 3 + 0] = pc[i * 4 + 0];
    xyz0[i * 3 + 1] = pc[i * 4 + 1];
    xyz0[i * 3 + 2] = pc[i * 4 + 2];
    feat0[i] = pc[i * 4 + 3];
    hm0[i] = hmi[i];
}

// ------------------------------------------------- weighted FPS (1 block, persistent)
__global__ void wfps_kernel(const float* __restrict__ xyz, const float* __restrict__ hm,
                            float* __restrict__ dist, int* __restrict__ inds,
                            int N, int npoint) {
    __shared__ float sv[1024];
    __shared__ int   si[1024];
    const int tid = threadIdx.x;
    const int nth = blockDim.x;
    for (int i = tid; i < N; i += nth) dist[i] = 1e10f;
    __syncthreads();
    int far = 0;
    for (int it = 0; it < npoint; ++it) {
        if (tid == 0) inds[it] = far;
        const float cx = xyz[far * 3 + 0];
        const float cy = xyz[far * 3 + 1];
        const float cz = xyz[far * 3 + 2];
        float best = -1.0f; int bidx = 0;
        for (int i = tid; i < N; i += nth) {
            float dx = xyz[i * 3 + 0] - cx;
            float dy = xyz[i * 3 + 1] - cy;
            float dz = xyz[i * 3 + 2] - cz;
            float d = dx * dx + dy * dy + dz * dz;
            float dm = fminf(dist[i], d);
            dist[i] = dm;
            float w = hm[i];
            float s = dm * w * w;          // alpha = 2
            if (s > best) { best = s; bidx = i; }
        }
        sv[tid] = best; si[tid] = bidx;
        __syncthreads();
        for (int stride = nth >> 1; stride > 0; stride >>= 1) {
            if (tid < stride) {
                float ov = sv[tid + stride]; int oi = si[tid + stride];
                if (ov > sv[tid] || (ov == sv[tid] && oi < si[tid])) { sv[tid] = ov; si[tid] = oi; }
            }
            __syncthreads();
        }
        far = si[0];
        __syncthreads();
    }
}

// ---------------------------------------------------------------- gather centroids + heat
__global__ void gather_pts_kernel(const float* __restrict__ xyz, const float* __restrict__ hm,
                                  const int* __restrict__ inds, float* __restrict__ nxyz,
                                  float* __restrict__ hmo, int S) {
    int s = blockIdx.x * blockDim.x + threadIdx.x;
    if (s >= S) return;
    int id = inds[s];
    nxyz[s * 3 + 0] = xyz[id * 3 + 0];
    nxyz[s * 3 + 1] = xyz[id * 3 + 1];
    nxyz[s * 3 + 2] = xyz[id * 3 + 2];
    if (hmo) hmo[s] = hm[id];
}

// ---------------------------------------------------------------- ball query
__global__ void ball_query_kernel(const float* __restrict__ xyz, const float* __restrict__ nxyz,
                                  int* __restrict__ idx, int N, int S, int ns, float r2) {
    int s = blockIdx.x * blockDim.x + threadIdx.x;
    if (s >= S) return;
    float cx = nxyz[s * 3 + 0], cy = nxyz[s * 3 + 1], cz = nxyz[s * 3 + 2];
    int cnt = 0, first = 0;
    for (int i = 0; i < N && cnt < ns; ++i) {
        float dx = xyz[i * 3 + 0] - cx;
        float dy = xyz[i * 3 + 1] - cy;
        float dz = xyz[i * 3 + 2] - cz;
        float d2 = dx * dx + dy * dy + dz * dz;
        if (d2 < r2) {
            if (cnt == 0) first = i;
            idx[(size_t)s * ns + cnt] = i;
            ++cnt;
        }
    }
    for (int j = cnt; j < ns; ++j) idx[(size_t)s * ns + j] = first;
}

// ---------------------------------------------------------------- SA input build
__global__ void build_rel_xyz_kernel(const float* __restrict__ xyz, const float* __restrict__ nxyz,
                                     const int* __restrict__ bidx, float* __restrict__ X,
                                     float invr, int K, int M, int s0) {
    int t = blockIdx.x * blockDim.x + threadIdx.x;
    if (t >= M) return;
    int sl = t / K, k = t - sl * K;
    int s = s0 + sl;
    int id = bidx[(size_t)s * K + k];
    float nx = nxyz[s * 3 + 0], ny = nxyz[s * 3 + 1], nz = nxyz[s * 3 + 2];
    X[t]                 = (xyz[id * 3 + 0] - nx) * invr;
    X[(size_t)M + t]     = (xyz[id * 3 + 1] - ny) * invr;
    X[(size_t)2 * M + t] = (xyz[id * 3 + 2] - nz) * invr;
}

__global__ void gather_feat_kernel(const float* __restrict__ feats, int C, int Npts,
                                   const int* __restrict__ bidx, float* __restrict__ X,
                                   int K, int M, int s0) {
    long u = (long)blockIdx.x * blockDim.x + threadIdx.x;
    if (u >= (long)C * M) return;
    int c = (int)(u / M);
    int t = (int)(u - (long)c * M);
    int sl = t / K, k = t - sl * K;
    int id = bidx[(size_t)(s0 + sl) * K + k];
    X[(size_t)(3 + c) * M + t] = feats[(size_t)c * Npts + id];
}

// ---------------------------------------------------------------- WMMA GEMM + ReLU
// Y[O][M] = relu(W[O][Cin] @ X[Cin][M]); one 16x16 output tile per wave32.
// Main K-loop is guard-free (no lane-dependent branches -> no EXEC masking);
// ragged Cin (131, 259) handled by one guarded tail step.
__global__ void gemm_relu_wmma_kernel(const float* __restrict__ W, const float* __restrict__ X,
                                      float* __restrict__ Y, int O, int Cin, int M, int doRelu) {
    int lane = threadIdx.x & 31;
    int wave = blockIdx.x * (blockDim.x >> 5) + (threadIdx.x >> 5);
    int mTiles = M >> 4;
    int total = (O >> 4) * mTiles;
    if (wave >= total) return;                 // wave-uniform: EXEC stays all-ones for actives
    int ot = wave / mTiles;
    int mt = wave - ot * mTiles;
    int oBase = ot << 4, mBase = mt << 4;
    int half = lane >> 4;                      // 0: K0/K1 slice, 1: K2/K3 slice
    int lr = lane & 15;
    const float* Wrow = W + (size_t)(oBase + lr) * Cin + (half << 1);
    const float* Xcol = X + (size_t)(half << 1) * M + (mBase + lr);
    v8f acc = {0.f, 0.f, 0.f, 0.f, 0.f, 0.f, 0.f, 0.f};
    const int cMain = Cin & ~3;
#pragma unroll 2
    for (int kk = 0; kk < cMain; kk += 4) {
        v2f a, b;
        a.x = Wrow[kk];                          // contiguous pair -> b64 load
        a.y = Wrow[kk + 1];
        b.x = Xcol[(size_t)kk * M];
        b.y = Xcol[(size_t)(kk + 1) * M];
        acc = __builtin_amdgcn_wmma_f32_16x16x4_f32(false, a, false, b, (short)0, acc, false, false);
    }
    if (cMain < Cin) {                           // single ragged tail step
        int ka = cMain + (half << 1);
        v2f a, b;
        a.x = (ka     < Cin) ? Wrow[cMain]     : 0.0f;
        a.y = (ka + 1 < Cin) ? Wrow[cMain + 1] : 0.0f;
        b.x = (ka     < Cin) ? Xcol[(size_t)cMain * M]       : 0.0f;
        b.y = (ka + 1 < Cin) ? Xcol[(size_t)(cMain + 1) * M] : 0.0f;
        acc = __builtin_amdgcn_wmma_f32_16x16x4_f32(false, a, false, b, (short)0, acc, false, false);
    }
    int col = mBase + lr;
#pragma unroll
    for (int v = 0; v < 8; ++v) {
        int row = oBase + v + (half << 3);     // lanes 0-15: M=v, lanes 16-31: M=v+8
        float val = acc[v];
        if (doRelu) val = fmaxf(val, 0.0f);
        Y[(size_t)row * M + col] = val;
    }
}

// ---------------------------------------------------------------- max over neighbors
__global__ void maxpool_kernel(const float* __restrict__ X, float* __restrict__ out,
                               int O, int Sc, int K, int s0, int Stot) {
    int t = blockIdx.x * blockDim.x + threadIdx.x;
    if (t >= O * Sc) return;
    int o = t / Sc, sl = t - o * Sc;
    const float* p = X + (size_t)o * Sc * K + (size_t)sl * K;
    float m = p[0];
    for (int k = 1; k < K; ++k) m = fmaxf(m, p[k]);
    out[(size_t)o * Stot + s0 + sl] = m;
}

// ---------------------------------------------------------------- 3-NN + weights
__global__ void knn3_kernel(const float* __restrict__ unk, const float* __restrict__ kn,
                            int Nu, int Nk, int* __restrict__ idx3, float* __restrict__ w3) {
    int n = blockIdx.x * blockDim.x + threadIdx.x;
    if (n >= Nu) return;
    float ux = unk[n * 3 + 0], uy = unk[n * 3 + 1], uz = unk[n * 3 + 2];
    float d0 = 3.4e38f, d1 = 3.4e38f, d2 = 3.4e38f;
    int i0 = 0, i1 = 0, i2 = 0;
    for (int i = 0; i < Nk; ++i) {
        float dx = kn[i * 3 + 0] - ux;
        float dy = kn[i * 3 + 1] - uy;
        float dz = kn[i * 3 + 2] - uz;
        float d = dx * dx + dy * dy + dz * dz;
        if (d < d0)      { d2 = d1; i2 = i1; d1 = d0; i1 = i0; d0 = d; i0 = i; }
        else if (d < d1) { d2 = d1; i2 = i1; d1 = d; i1 = i; }
        else if (d < d2) { d2 = d; i2 = i; }
    }
    float r0 = 1.0f / (d0 + 1e-8f);
    float r1 = 1.0f / (d1 + 1e-8f);
    float r2v = 1.0f / (d2 + 1e-8f);
    float s = r0 + r1 + r2v;
    idx3[n * 3 + 0] = i0; idx3[n * 3 + 1] = i1; idx3[n * 3 + 2] = i2;
    w3[n * 3 + 0] = r0 / s; w3[n * 3 + 1] = r1 / s; w3[n * 3 + 2] = r2v / s;
}

// ---------------------------------------------------------------- FP input: [interp ; skip]
__global__ void build_fp_input_kernel(const float* __restrict__ kf, const float* __restrict__ uf,
                                      const int* __restrict__ idx3, const float* __restrict__ w3,
                                      float* __restrict__ X, int Ck, int Cu, int Nu, int Nk) {
    long t = (long)blockIdx.x * blockDim.x + threadIdx.x;
    if (t >= (long)(Ck + Cu) * Nu) return;
    int c = (int)(t / Nu);
    int n = (int)(t - (long)c * Nu);
    if (c < Ck) {
        const float* row = kf + (size_t)c * Nk;
        X[t] = row[idx3[n * 3 + 0]] * w3[n * 3 + 0]
             + row[idx3[n * 3 + 1]] * w3[n * 3 + 1]
             + row[idx3[n * 3 + 2]] * w3[n * 3 + 2];
    } else {
        X[t] = uf[(size_t)(c - Ck) * Nu + n];
    }
}

// ================================================================ host
extern "C" void kernel_launch(void* const* d_in, const int* in_sizes, int n_in,
                              void* d_out, int out_size, void* d_ws, size_t ws_size,
                              hipStream_t stream) {
    (void)in_sizes; (void)n_in; (void)out_size; (void)ws_size;
    const float* pc   = (const float*)d_in[0];
    const float* hmi  = (const float*)d_in[1];
    const float* h1x  = (const float*)d_in[2];
    const float* h1f  = (const float*)d_in[3];
    const float* h2x  = (const float*)d_in[4];
    const float* h2f  = (const float*)d_in[5];
    const float* h3x  = (const float*)d_in[6];
    const float* h3f  = (const float*)d_in[7];
    const float* h4x  = (const float*)d_in[8];
    const float* h4f  = (const float*)d_in[9];
    const float* sa1w0 = (const float*)d_in[10];
    const float* sa1w1 = (const float*)d_in[11];
    const float* sa1w2 = (const float*)d_in[12];
    const float* sa2w0 = (const float*)d_in[13];
    const float* sa2w1 = (const float*)d_in[14];
    const float* sa2w2 = (const float*)d_in[15];
    const float* sa3w0 = (const float*)d_in[16];
    const float* sa3w1 = (const float*)d_in[17];
    const float* sa3w2 = (const float*)d_in[18];
    const float* sa4w0 = (const float*)d_in[19];
    const float* sa4w1 = (const float*)d_in[20];
    const float* sa4w2 = (const float*)d_in[21];
    const float* fp1fw = (const float*)d_in[22];
    const float* fp2fw = (const float*)d_in[23];
    const float* fp3fw = (const float*)d_in[24];
    const float* fp4fw = (const float*)d_in[25];
    const float* fp1w0 = (const float*)d_in[26];
    const float* fp1w1 = (const float*)d_in[27];
    const float* fp2w0 = (const float*)d_in[28];
    const float* fp2w1 = (const float*)d_in[29];

    // ------- workspace carve (aligned 256B)
    char* base = (char*)d_ws;
    size_t off = 0;
    auto alloc = [&](size_t elems, size_t esz) -> void* {
        off = (off + 255) & ~(size_t)255;
        void* p = base + off;
        off += elems * esz;
        return p;
    };
    const int N0 = 20000;
    float* xyz0  = (float*)alloc(N0 * 3, 4);
    float* feat0 = (float*)alloc(N0, 4);
    float* hm0   = (float*)alloc(N0, 4);
    float* hm1   = (float*)alloc(4096, 4);
    float* hm2   = (float*)alloc(2048, 4);
    float* hm3   = (float*)alloc(1024, 4);
    float* dist  = (float*)alloc(N0, 4);
    int*   inds  = (int*)alloc(4096, 4);
    float* nx1   = (float*)alloc(4096 * 3, 4);
    float* nx2   = (float*)alloc(2048 * 3, 4);
    float* nx3   = (float*)alloc(1024 * 3, 4);
    float* nx4   = (float*)alloc(512 * 3, 4);
    int*   bidx  = (int*)alloc(4096 * 64, 4);
    int*   kidx  = (int*)alloc(4096 * 3, 4);
    float* kw    = (float*)alloc(4096 * 3, 4);
    float* f1p   = (float*)alloc(128 * 4096, 4);
    float* f1    = (float*)alloc(128 * 4096, 4);
    float* f2p   = (float*)alloc(256 * 2048, 4);
    float* f2    = (float*)alloc(256 * 2048, 4);
    float* f3p   = (float*)alloc(256 * 1024, 4);
    float* f3    = (float*)alloc(256 * 1024, 4);
    float* f3b   = (float*)alloc(256 * 1024, 4);
    float* f4p   = (float*)alloc(256 * 512, 4);
    float* f4    = (float*)alloc(256 * 512, 4);
    float* XA    = (float*)alloc((size_t)4194304, 4);   // 16 MB ping
    float* XB    = (float*)alloc((size_t)4194304, 4);   // 16 MB pong

    auto gb = [](long n) { return dim3((unsigned)((n + TPB - 1) / TPB)); };

    auto gemm = [&](const float* W, const float* X, float* Y, int O, int Cin, int M) {
        long waves = (long)(O / 16) * (M / 16);
        gemm_relu_wmma_kernel<<<gb(waves * 32), TPB, 0, stream>>>(W, X, Y, O, Cin, M, 1);
    };

    auto run_sa = [&](const float* xyzS, int Np, const float* featS, int C,
                      const float* nx, int NP, int K, float r,
                      const float* W0, int O0, const float* W1, int O1,
                      const float* W2, int O2, float* fpre) {
        const int CH = 512;
        for (int s0 = 0; s0 < NP; s0 += CH) {
            int Sc = (NP - s0 < CH) ? (NP - s0) : CH;
            int M = Sc * K;
            build_rel_xyz_kernel<<<gb(M), TPB, 0, stream>>>(xyzS, nx, bidx, XA, 1.0f / r, K, M, s0);
            gather_feat_kernel<<<gb((long)C * M), TPB, 0, stream>>>(featS, C, Np, bidx, XA, K, M, s0);
            gemm(W0, XA, XB, O0, 3 + C, M);
            gemm(W1, XB, XA, O1, O0, M);
            gemm(W2, XA, XB, O2, O1, M);
            maxpool_kernel<<<gb((long)O2 * Sc), TPB, 0, stream>>>(XB, fpre, O2, Sc, K, s0, NP);
        }
    };

    // ---------------- stage 0: unpack
    prep_kernel<<<gb(N0), TPB, 0, stream>>>(pc, hmi, xyz0, feat0, hm0, N0);

    // ---------------- level 1
    wfps_kernel<<<1, 1024, 0, stream>>>(xyz0, hm0, dist, inds, N0, 4096);
    gather_pts_kernel<<<gb(4096), TPB, 0, stream>>>(xyz0, hm0, inds, nx1, hm1, 4096);
    ball_query_kernel<<<gb(4096), TPB, 0, stream>>>(xyz0, nx1, bidx, N0, 4096, 64, 0.04f * 0.04f);
    run_sa(xyz0, N0, feat0, 1, nx1, 4096, 64, 0.04f, sa1w0, 64, sa1w1, 64, sa1w2, 128, f1p);
    knn3_kernel<<<gb(4096), TPB, 0, stream>>>(nx1, h1x, 4096, 4096, kidx, kw);
    build_fp_input_kernel<<<gb((long)256 * 4096), TPB, 0, stream>>>(h1f, f1p, kidx, kw, XA, 128, 128, 4096, 4096);
    gemm(fp1fw, XA, f1, 128, 256, 4096);

    // ---------------- level 2
    wfps_kernel<<<1, 1024, 0, stream>>>(nx1, hm1, dist, inds, 4096, 2048);
    gather_pts_kernel<<<gb(2048), TPB, 0, stream>>>(nx1, hm1, inds, nx2, hm2, 2048);
    ball_query_kernel<<<gb(2048), TPB, 0, stream>>>(nx1, nx2, bidx, 4096, 2048, 32, 0.08f * 0.08f);
    run_sa(nx1, 4096, f1, 128, nx2, 2048, 32, 0.08f, sa2w0, 128, sa2w1, 128, sa2w2, 256, f2p);
    knn3_kernel<<<gb(2048), TPB, 0, stream>>>(nx2, h2x, 2048, 2048, kidx, kw);
    build_fp_input_kernel<<<gb((long)512 * 2048), TPB, 0, stream>>>(h2f, f2p, kidx, kw, XA, 256, 256, 2048, 2048);
    gemm(fp2fw, XA, f2, 256, 512, 2048);

    // ---------------- level 3
    wfps_kernel<<<1, 1024, 0, stream>>>(nx2, hm2, dist, inds, 2048, 1024);
    gather_pts_kernel<<<gb(1024), TPB, 0, stream>>>(nx2, hm2, inds, nx3, hm3, 1024);
    ball_query_kernel<<<gb(1024), TPB, 0, stream>>>(nx2, nx3, bidx, 2048, 1024, 16, 0.16f * 0.16f);
    run_sa(nx2, 2048, f2, 256, nx3, 1024, 16, 0.16f, sa3w0, 128, sa3w1, 128, sa3w2, 256, f3p);
    knn3_kernel<<<gb(1024), TPB, 0, stream>>>(nx3, h3x, 1024, 1024, kidx, kw);
    build_fp_input_kernel<<<gb((long)512 * 1024), TPB, 0, stream>>>(h3f, f3p, kidx, kw, XA, 256, 256, 1024, 1024);
    gemm(fp3fw, XA, f3, 256, 512, 1024);

    // ---------------- level 4
    wfps_kernel<<<1, 1024, 0, stream>>>(nx3, hm3, dist, inds, 1024, 512);
    gather_pts_kernel<<<gb(512), TPB, 0, stream>>>(nx3, hm3, inds, nx4, (float*)nullptr, 512);
    ball_query_kernel<<<gb(512), TPB, 0, stream>>>(nx3, nx4, bidx, 1024, 512, 16, 0.24f * 0.24f);
    run_sa(nx3, 1024, f3, 256, nx4, 512, 16, 0.24f, sa4w0, 128, sa4w1, 128, sa4w2, 256, f4p);
    knn3_kernel<<<gb(512), TPB, 0, stream>>>(nx4, h4x, 512, 512, kidx, kw);
    build_fp_input_kernel<<<gb((long)512 * 512), TPB, 0, stream>>>(h4f, f4p, kidx, kw, XA, 256, 256, 512, 512);
    gemm(fp4fw, XA, f4, 256, 512, 512);

    // ---------------- fp1: xyz3 <- xyz4
    knn3_kernel<<<gb(1024), TPB, 0, stream>>>(nx3, nx4, 1024, 512, kidx, kw);
    build_fp_input_kernel<<<gb((long)512 * 1024), TPB, 0, stream>>>(f4, f3, kidx, kw, XA, 256, 256, 1024, 512);
    gemm(fp1w0, XA, XB, 256, 512, 1024);
    gemm(fp1w1, XB, f3b, 256, 256, 1024);

    // ---------------- fp2: xyz2 <- xyz3  -> output
    knn3_kernel<<<gb(2048), TPB, 0, stream>>>(nx2, nx3, 2048, 1024, kidx, kw);
    build_fp_input_kernel<<<gb((long)512 * 2048), TPB, 0, stream>>>(f3b, f2, kidx, kw, XA, 256, 256, 2048, 1024);
    gemm(fp2w0, XA, XB, 256, 512, 2048);
    gemm(fp2w1, XB, (float*)d_out, 256, 256, 2048);
}